// FastestDet_38465727103749
// MI455X (gfx1250) — compile-verified
//
#include <hip/hip_runtime.h>
#include <hip/hip_bf16.h>
#include <math.h>

#define CONF_TH 0.25f
#define NMS_TH  0.35f
#define N_IMG   64
#define NCH     85
#define HW      1024   // 32*32
#define NCLS    80

typedef __attribute__((ext_vector_type(2))) float        v2f;
typedef __attribute__((ext_vector_type(8))) float        v8f;
typedef __attribute__((ext_vector_type(4))) unsigned int u32x4;
typedef __attribute__((ext_vector_type(8))) int          i32x8;
typedef __attribute__((ext_vector_type(4))) int          i32x4;

// ---------------------------------------------------------------------------
// Kernel 1: decode [N,85,32,32] -> [N,1024,6] (x1,y1,x2,y2,score,cls)
// ---------------------------------------------------------------------------
__global__ __launch_bounds__(256) void fd_decode(const float* __restrict__ preds,
                                                 float* __restrict__ boxes) {
    const int gid  = blockIdx.x * 256 + threadIdx.x;   // 0..65535
    const int n    = gid >> 10;
    const int cell = gid & 1023;
    const int h    = cell >> 5;
    const int w    = cell & 31;

    const float* p = preds + (size_t)n * NCH * HW + cell;   // channel stride = HW
    const float pobj = p[0];
    const float tx = p[1 * HW], ty = p[2 * HW], tw = p[3 * HW], th = p[4 * HW];

    float maxv = -__builtin_inff();
    int   maxi = 0;
    #pragma unroll 8
    for (int c = 0; c < NCLS; ++c) {
        const float v = p[(5 + c) * HW];
        if (v > maxv) { maxv = v; maxi = c; }    // strict > => first index on ties
    }
    const float score = pobj * maxv;
    const float bw  = 1.0f / (1.0f + expf(-tw));  // sigmoid
    const float bh  = 1.0f / (1.0f + expf(-th));
    const float bcx = (tanhf(tx) + (float)w) * (1.0f / 32.0f);
    const float bcy = (tanhf(ty) + (float)h) * (1.0f / 32.0f);

    float* o = boxes + (size_t)gid * 6;
    o[0] = bcx - 0.5f * bw;
    o[1] = bcy - 0.5f * bh;
    o[2] = bcx + 0.5f * bw;
    o[3] = bcy + 0.5f * bh;
    o[4] = score;
    o[5] = (float)maxi;
}

// ---------------------------------------------------------------------------
// Kernel 2: suppression bitmatrix (original-box order; symmetric predicate)
//   mat[img][i][word j/32] bit (j%32) = (IoU(i,j) > NMS_TH) && same_cls
// One wave32 per 16x32 tile; area_i+area_j via v_wmma_f32_16x16x4_f32:
//   A(16x4): lane m<16: {area_row[m], 1.0}, lanes>=16 (K=2,3): {0,0}
//   B(4x16): VGPR0 lanes0-15 = row K0 = 1.0 ; lanes16-31 = row K1 = area_col[n]
//   D(16x16): VGPR r = row r (lanes 0-15) / row r+8 (lanes 16-31), col = lane&15
// IoU test is division-free: denom > 0 (areas > 0), so
//   inter/denom > TH  <=>  inter > TH*denom.
// Every 32-bit mask word written exactly once -> no atomics, no pre-zero.
// ---------------------------------------------------------------------------
__global__ __launch_bounds__(256) void fd_suppmat(const float* __restrict__ boxes,
                                                  unsigned int* __restrict__ mat) {
    const int img  = blockIdx.y;
    const int wave = threadIdx.x >> 5;
    const int lane = threadIdx.x & 31;
    const int tile = blockIdx.x * 8 + wave;    // 0..2047 per image
    const int rowTile = tile >> 5;             // 0..63
    const int colTile = tile & 31;             // 0..31
    const int R = rowTile << 4;                // 16 rows
    const int C = colTile << 5;                // 32 cols

    __shared__ float s_row[8][16][5];          // x1,y1,x2,y2,cls
    __shared__ float s_rarea[8][16];

    const float* ib = boxes + (size_t)img * HW * 6;
    if (lane < 16) {
        const float* rb = ib + (size_t)(R + lane) * 6;
        const float x1 = rb[0], y1 = rb[1], x2 = rb[2], y2 = rb[3];
        s_row[wave][lane][0] = x1;
        s_row[wave][lane][1] = y1;
        s_row[wave][lane][2] = x2;
        s_row[wave][lane][3] = y2;
        s_row[wave][lane][4] = rb[5];
        s_rarea[wave][lane]  = (x2 - x1) * (y2 - y1);
    }
    __syncthreads();

    // Per-lane column boxes for the two 16-col subtiles.
    const int   n   = lane & 15;
    const float* ca = ib + (size_t)(C + n) * 6;
    const float* cb = ib + (size_t)(C + 16 + n) * 6;
    const float ax1 = ca[0], ay1 = ca[1], ax2 = ca[2], ay2 = ca[3], acl = ca[5];
    const float bx1 = cb[0], by1 = cb[1], bx2 = cb[2], by2 = cb[3], bcl = cb[5];
    const float aArea = (ax2 - ax1) * (ay2 - ay1);
    const float bArea = (bx2 - bx1) * (by2 - by1);

    v2f A;  A.x  = (lane < 16) ? s_rarea[wave][lane] : 0.0f;
            A.y  = (lane < 16) ? 1.0f : 0.0f;
    v2f B1; B1.x = (lane < 16) ? 1.0f : aArea;  B1.y = 0.0f;
    v2f B2; B2.x = (lane < 16) ? 1.0f : bArea;  B2.y = 0.0f;
    v8f cz = {};
    // D[m][n] = area_row[m] + area_col[n]
    v8f d1 = __builtin_amdgcn_wmma_f32_16x16x4_f32(false, A, false, B1, (short)0, cz, false, false);
    v8f d2 = __builtin_amdgcn_wmma_f32_16x16x4_f32(false, A, false, B2, (short)0, cz, false, false);

    const int hi = lane >> 4;                  // 0: rows 0-7, 1: rows 8-15
    unsigned int mA[8], mB[8];
    #pragma unroll
    for (int r = 0; r < 8; ++r) {
        const int ri = hi * 8 + r;
        const float rx1 = s_row[wave][ri][0], ry1 = s_row[wave][ri][1];
        const float rx2 = s_row[wave][ri][2], ry2 = s_row[wave][ri][3];
        const float rcl = s_row[wave][ri][4];

        float iw = fmaxf(fminf(rx2, ax2) - fmaxf(rx1, ax1), 0.0f);
        float ih = fmaxf(fminf(ry2, ay2) - fmaxf(ry1, ay1), 0.0f);
        float inter = iw * ih;
        // inter/(sum-inter+eps) > TH  <=>  inter > TH*(sum-inter+eps)  (denom>0)
        bool sA = (inter > NMS_TH * (d1[r] - inter + 1e-9f)) && (rcl == acl);
        mA[r] = (unsigned int)__ballot(sA);

        iw = fmaxf(fminf(rx2, bx2) - fmaxf(rx1, bx1), 0.0f);
        ih = fmaxf(fminf(ry2, by2) - fmaxf(ry1, by1), 0.0f);
        inter = iw * ih;
        bool sB = (inter > NMS_TH * (d2[r] - inter + 1e-9f)) && (rcl == bcl);
        mB[r] = (unsigned int)__ballot(sB);
    }

    if (lane == 0) {
        unsigned int* base = mat + ((size_t)img << 15) + (unsigned)colTile;
        #pragma unroll
        for (int k = 0; k < 8; ++k) {
            // mA/mB bits0-15 = row k, bits16-31 = row k+8 (cols lo/hi 16)
            const unsigned int wlo = (mA[k] & 0xFFFFu) | ((mB[k] & 0xFFFFu) << 16);
            const unsigned int whi = (mA[k] >> 16)     | (mB[k] & 0xFFFF0000u);
            base[(size_t)(R + k)     * 32] = wlo;
            base[(size_t)(R + 8 + k) * 32] = whi;
        }
    }
}

// ---------------------------------------------------------------------------
// Kernel 3: per-image bitonic sort + sequential greedy NMS.
// Boxes staged into LDS with the Tensor Data Mover (tensor_load_to_lds).
// ---------------------------------------------------------------------------
__global__ __launch_bounds__(1024) void fd_nms(const float* __restrict__ boxes,
                                               const unsigned int* __restrict__ mat,
                                               float* __restrict__ keep_out) {
    const int img = blockIdx.x;
    const int tid = threadIdx.x;

    __shared__ float        s_boxes[HW * 6];   // 24 KB
    __shared__ float        s_key[HW];
    __shared__ int          s_idx[HW];
    __shared__ unsigned int s_rowv[32];
    __shared__ int          s_supp;

    // --- TDM: async DMA of this image's 1024x6 f32 tile into LDS (wave 0) ---
    const float* src = boxes + (size_t)img * HW * 6;
    if (tid < 32) {
        const unsigned long long ga   = (unsigned long long)(uintptr_t)src;
        const unsigned int       ldsa = (unsigned int)(uintptr_t)(&s_boxes[0]);
        const unsigned int       el   = HW * 6;   // 6144 elements of 4 bytes
        // D# group0: count=1, lds_addr, global_addr[56:0], type=2
        u32x4 g0 = { 1u,
                     ldsa,
                     (unsigned int)(ga & 0xFFFFFFFFull),
                     (unsigned int)((ga >> 32) & 0x01FFFFFFull) | (2u << 30) };
        // D# group1: data_size=4B; tensor_dim0=6144, tensor_dim1=1,
        //            tile_dim0=6144, tile_dim1=1, tensor_dim0_stride=6144
        i32x8 g1 = { (int)(2u << 16),
                     (int)((el & 0xFFFFu) << 16),
                     (int)(((el >> 16) & 0xFFFFu) | (1u << 16)),
                     (int)((el & 0xFFFFu) << 16),
                     1,
                     (int)el,
                     0, 0 };
        i32x4 gz4 = { 0, 0, 0, 0 };
        i32x8 gz8 = { 0, 0, 0, 0, 0, 0, 0, 0 };
        // 6-arg toolchain variant: (g0, g1, g2, g3, g4, cpol)
        __builtin_amdgcn_tensor_load_to_lds(g0, g1, gz4, gz4, gz8, 0);
        __builtin_amdgcn_s_wait_tensorcnt(0);
    }
    __syncthreads();

    // --- bitonic sort ascending on key = -score (i.e., descending score) ---
    s_key[tid] = -s_boxes[tid * 6 + 4];
    s_idx[tid] = tid;
    __syncthreads();
    for (int k = 2; k <= HW; k <<= 1) {
        for (int j = k >> 1; j > 0; j >>= 1) {
            const int ixj = tid ^ j;
            if (ixj > tid) {
                const bool  up = ((tid & k) == 0);
                const float a = s_key[tid], b = s_key[ixj];
                if (up ? (a > b) : (a < b)) {
                    s_key[tid] = b; s_key[ixj] = a;
                    const int t = s_idx[tid]; s_idx[tid] = s_idx[ixj]; s_idx[ixj] = t;
                }
            }
            __syncthreads();
        }
    }

    // --- greedy scan: thread tid owns sorted position tid ---
    const float myScore = -s_key[tid];
    const int   myO     = s_idx[tid];
    const bool  myValid = myScore > CONF_TH;
    bool        myKeep  = false;

    const unsigned int* imat = mat + ((size_t)img << 15);
    for (int i = 0; i < HW; ++i) {
        const int oi = s_idx[i];
        if (tid < 32) s_rowv[tid] = imat[(size_t)oi * 32 + tid];
        if (tid == 0) s_supp = 0;
        __syncthreads();
        if (myKeep && ((s_rowv[myO >> 5] >> (myO & 31)) & 1u)) s_supp = 1;
        __syncthreads();
        if (tid == i) myKeep = myValid && (s_supp == 0);
        __syncthreads();
    }
    keep_out[(size_t)img * HW + myO] = myKeep ? 1.0f : 0.0f;
}

// ---------------------------------------------------------------------------
extern "C" void kernel_launch(void* const* d_in, const int* in_sizes, int n_in,
                              void* d_out, int out_size, void* d_ws, size_t ws_size,
                              hipStream_t stream) {
    const float* preds = (const float*)d_in[0];
    float* out   = (float*)d_out;
    float* bxs   = out;                            // [64,1024,6]
    float* keep  = out + (size_t)N_IMG * HW * 6;   // [64,1024]
    unsigned int* mat = (unsigned int*)d_ws;       // 64 * 1024 * 32 words = 8 MB

    fd_decode <<<dim3(N_IMG * HW / 256), dim3(256), 0, stream>>>(preds, bxs);
    fd_suppmat<<<dim3(256, N_IMG),       dim3(256), 0, stream>>>(bxs, mat);
    fd_nms    <<<dim3(N_IMG),            dim3(1024), 0, stream>>>(bxs, mat, keep);
}